// New_Loss_82171314307698
// MI455X (gfx1250) — compile-verified
//
#include <hip/hip_runtime.h>
#include <hip/hip_bf16.h>

// ---------------------------------------------------------------------------
// mean_i max(0, 0.1 - (a[i] - b[i])) over N = 2^25 fp32 elements.
// Memory-bound: 256 MB read @ 23.3 TB/s => ~11 us floor. Strategy:
//   - b128 nontemporal streaming loads (data > 192 MB L2, single use)
//   - per-wave reduction via V_WMMA_F32_16X16X4_F32 (ones-matrix trick)
//   - two-pass deterministic reduction through d_ws
// ---------------------------------------------------------------------------

typedef __attribute__((ext_vector_type(2))) float v2f;
typedef __attribute__((ext_vector_type(4))) float v4f;
typedef __attribute__((ext_vector_type(8))) float v8f;

constexpr int   N_TOTAL = 33554432;          // 2^25
constexpr float MARGIN  = 0.1f;
constexpr int   BLOCK   = 256;               // 8 wave32s
constexpr int   ITERS   = 16;                // float4 iterations per thread
constexpr int   EPB     = BLOCK * ITERS * 4; // 16384 elements per block
constexpr int   NBLOCKS = N_TOTAL / EPB;     // 2048 blocks

// Wave32 all-lane sum using the matrix pipe.
// A[m][k]: lanes 0-15 supply (m=lane, k=0/1), lanes 16-31 supply (m=lane-16, k=2/3).
// With a = {p, 0} and B = ones(4x16):  D[m][n] = p_m + p_{m+16}  (row-constant).
// C layout: lane l<16 holds D[v][l] in c[v]; lane l>=16 holds D[v+8][l-16].
// Sum of 8 C regs -> half-wave sums; ds_swizzle SWAPX16 + add -> full wave sum.
__device__ __forceinline__ float wave_reduce_sum_wmma(float p) {
    v2f a; a.x = p;     a.y = 0.0f;
    v2f b; b.x = 1.0f;  b.y = 1.0f;
    v8f c = {};
    c = __builtin_amdgcn_wmma_f32_16x16x4_f32(
            /*neg_a=*/false, a, /*neg_b=*/false, b,
            /*c_mod=*/(short)0, c, /*reuse_a=*/false, /*reuse_b=*/false);
    float s = ((c[0] + c[1]) + (c[2] + c[3])) + ((c[4] + c[5]) + (c[6] + c[7]));
    // add the two 16-lane halves together: xor_mask=0x10, and_mask=0x1f -> 0x401F
    int other = __builtin_amdgcn_ds_swizzle(__float_as_int(s), 0x401F);
    return s + __int_as_float(other);
}

__global__ void __launch_bounds__(BLOCK)
hinge_partial_kernel(const float* __restrict__ amax,
                     const float* __restrict__ aorg,
                     float* __restrict__ ws) {
    const v4f* __restrict__ A = reinterpret_cast<const v4f*>(amax);
    const v4f* __restrict__ B = reinterpret_cast<const v4f*>(aorg);

    const int  tid  = threadIdx.x;
    const long base = (long)blockIdx.x * (EPB / 4) + tid;

    float acc0 = 0.0f, acc1 = 0.0f, acc2 = 0.0f, acc3 = 0.0f;
#pragma unroll
    for (int i = 0; i < ITERS; ++i) {
        v4f a = __builtin_nontemporal_load(&A[base + (long)i * BLOCK]);
        v4f b = __builtin_nontemporal_load(&B[base + (long)i * BLOCK]);
        acc0 += fmaxf(0.0f, MARGIN - (a.x - b.x));
        acc1 += fmaxf(0.0f, MARGIN - (a.y - b.y));
        acc2 += fmaxf(0.0f, MARGIN - (a.z - b.z));
        acc3 += fmaxf(0.0f, MARGIN - (a.w - b.w));
    }
    float p = (acc0 + acc1) + (acc2 + acc3);

    // Wave-level sum on the matrix pipe (EXEC all-1s here: no divergence above).
    float wsum = wave_reduce_sum_wmma(p);

    __shared__ float lds[BLOCK / 32];
    const int wave = tid >> 5;
    if ((tid & 31) == 0) lds[wave] = wsum;
    __syncthreads();

    if (tid == 0) {
        float t = 0.0f;
#pragma unroll
        for (int w = 0; w < BLOCK / 32; ++w) t += lds[w];
        ws[blockIdx.x] = t;
    }
}

__global__ void __launch_bounds__(BLOCK)
hinge_final_kernel(const float* __restrict__ ws, float* __restrict__ out) {
    const int tid = threadIdx.x;
    float p = 0.0f;
#pragma unroll
    for (int i = 0; i < NBLOCKS / BLOCK; ++i)   // 2048/256 = 8 per thread
        p += ws[tid + i * BLOCK];

    float wsum = wave_reduce_sum_wmma(p);

    __shared__ float lds[BLOCK / 32];
    const int wave = tid >> 5;
    if ((tid & 31) == 0) lds[wave] = wsum;
    __syncthreads();

    if (tid == 0) {
        float t = 0.0f;
#pragma unroll
        for (int w = 0; w < BLOCK / 32; ++w) t += lds[w];
        out[0] = t * (1.0f / (float)N_TOTAL);   // exact: 1/2^25
    }
}

extern "C" void kernel_launch(void* const* d_in, const int* in_sizes, int n_in,
                              void* d_out, int out_size, void* d_ws, size_t ws_size,
                              hipStream_t stream) {
    const float* amax = (const float*)d_in[0];
    const float* aorg = (const float*)d_in[1];
    float*       out  = (float*)d_out;
    float*       ws   = (float*)d_ws;   // needs NBLOCKS * 4 = 8 KB

    hinge_partial_kernel<<<NBLOCKS, BLOCK, 0, stream>>>(amax, aorg, ws);
    hinge_final_kernel<<<1, BLOCK, 0, stream>>>(ws, out);
}